// PointerNet_89395449299613
// MI455X (gfx1250) — compile-verified
//
#include <hip/hip_runtime.h>
#include <math.h>

#define S_LEN  1024
#define DMODEL 128
#define NHEAD  8
#define DHEAD  16
#define NLAYER 3
#define DFF    2048

typedef __attribute__((ext_vector_type(2))) float v2f;
typedef __attribute__((ext_vector_type(8))) float v8f;

// ---------------------------------------------------------------------------
// Embedding: xe[s][d] = x[s][:]@emb_w[d][:] + emb_b[d]   (B=1, seq-first)
// ---------------------------------------------------------------------------
__global__ void embed_kernel(const float* __restrict__ x,
                             const float* __restrict__ w,
                             const float* __restrict__ b,
                             float* __restrict__ xe) {
  int i = blockIdx.x * blockDim.x + threadIdx.x;   // s*128 + d
  int s = i >> 7, d = i & 127;
  xe[i] = x[s * 2 + 0] * w[d * 2 + 0] + x[s * 2 + 1] * w[d * 2 + 1] + b[d];
}

// ---------------------------------------------------------------------------
// One-shot transpose: out[c*rows + r] = in[r*ld + off + c]   (in is [rows,cols])
// Read-coalesced; runs once, so the strided write is irrelevant.
// ---------------------------------------------------------------------------
__global__ void transpose_kernel(const float* __restrict__ in,
                                 float* __restrict__ out,
                                 int rows, int cols, int ld, int off) {
  int i = blockIdx.x * blockDim.x + threadIdx.x;
  if (i >= rows * cols) return;
  int r = i / cols, c = i - r * cols;
  out[(size_t)c * rows + r] = in[(size_t)r * ld + off + c];
}

// ---------------------------------------------------------------------------
// C[M,N] = A[M,K] @ W[N,K]^T (+bias) (+relu)   using V_WMMA_F32_16X16X4_F32.
// One wave (32 lanes) owns one 16x16 output tile; grid = (N/16, M/16).
// f32 A-frag (16x4): lane m=lane&15; lanes<16 hold K={k,k+1}, lanes>=16 K={k+2,k+3}
// f32 B-frag (4x16): lane n=lane&15; same K split -> B[k][n] = W[n][k] (W row-major [N,K])
// C/D: VGPR r -> row (r + 8*(lane>>4)), col (lane&15).
// transC=1 stores C transposed ([N,M], row stride ldc=M): each lane then owns
// 8 consecutive elements -> contiguous per-lane store (merges to b128).
// ---------------------------------------------------------------------------
__global__ void gemm_wmma(const float* __restrict__ A, int lda,
                          const float* __restrict__ W, int ldw,
                          const float* __restrict__ bias,
                          float* __restrict__ C, int ldc,
                          int K, int relu, int transC) {
  const int lane = threadIdx.x;
  const int lo = lane & 15;
  const int hi = lane >> 4;
  const int tile_n = blockIdx.x;
  const int tile_m = blockIdx.y;

  const float* arow = A + (size_t)(tile_m * 16 + lo) * lda + 2 * hi;
  const float* wrow = W + (size_t)(tile_n * 16 + lo) * ldw + 2 * hi;

  v8f acc = {};
#if __has_builtin(__builtin_amdgcn_wmma_f32_16x16x4_f32)
  #pragma unroll 4
  for (int k = 0; k < K; k += 4) {
    v2f a = *(const v2f*)(arow + k);   // 8B-aligned: lda,k multiples of 4, 2*hi even
    v2f b = *(const v2f*)(wrow + k);
    acc = __builtin_amdgcn_wmma_f32_16x16x4_f32(
        /*neg_a=*/false, a, /*neg_b=*/false, b,
        /*c_mod=*/(short)0, acc, /*reuse_a=*/false, /*reuse_b=*/false);
  }
#else
  // Scalar fallback producing the same D layout (should not be taken on gfx1250).
  const float* ar2 = A + (size_t)(tile_m * 16 + 8 * hi) * lda;
  const float* wr2 = W + (size_t)(tile_n * 16 + lo) * ldw;
  for (int r = 0; r < 8; ++r) {
    float s = 0.0f;
    for (int k = 0; k < K; ++k) s += ar2[(size_t)r * lda + k] * wr2[k];
    acc[r] = s;
  }
#endif

  const int col = tile_n * 16 + lo;
  const float bv = bias ? bias[col] : 0.0f;
  const int rbase = tile_m * 16 + 8 * hi;
  if (transC) {
    float* cp = C + (size_t)col * ldc + rbase;   // 8 contiguous floats per lane
    #pragma unroll
    for (int r = 0; r < 8; ++r) {
      float v = acc[r] + bv;
      if (relu) v = fmaxf(v, 0.0f);
      cp[r] = v;
    }
  } else {
    #pragma unroll
    for (int r = 0; r < 8; ++r) {
      float v = acc[r] + bv;
      if (relu) v = fmaxf(v, 0.0f);
      C[(size_t)(rbase + r) * ldc + col] = v;
    }
  }
}

// ---------------------------------------------------------------------------
// Fused MHA (B=1): one thread per (query s, head h), online softmax over keys.
// qkv layout: [S, 3D] with q|k|v blocks; head h occupies dims h*16..h*16+15.
// Output ao: [S, D] (= concat of heads, matching the reference reshape).
// ---------------------------------------------------------------------------
__global__ void attn_kernel(const float* __restrict__ qkv,
                            float* __restrict__ ao) {
  int gid = blockIdx.x * blockDim.x + threadIdx.x;
  int s = gid >> 3, h = gid & 7;

  const float* qp = qkv + (size_t)s * (3 * DMODEL) + h * DHEAD;
  float q[DHEAD];
  #pragma unroll
  for (int i = 0; i < DHEAD; ++i) q[i] = qp[i] * 0.25f;   // 1/sqrt(16)

  float m = -__builtin_inff(), l = 0.0f;
  float av[DHEAD];
  #pragma unroll
  for (int i = 0; i < DHEAD; ++i) av[i] = 0.0f;

  for (int k = 0; k < S_LEN; ++k) {
    const float* kp = qkv + (size_t)k * (3 * DMODEL) + DMODEL + h * DHEAD;
    const float* vp = kp + DMODEL;
    float sc = 0.0f;
    #pragma unroll
    for (int i = 0; i < DHEAD; ++i) sc += q[i] * kp[i];
    float mn = fmaxf(m, sc);
    float corr = __expf(m - mn);          // exp(-inf)=0 on first key
    float p = __expf(sc - mn);
    l = l * corr + p;
    #pragma unroll
    for (int i = 0; i < DHEAD; ++i) av[i] = av[i] * corr + p * vp[i];
    m = mn;
  }
  float inv = 1.0f / l;
  float* op = ao + (size_t)s * DMODEL + h * DHEAD;
  #pragma unroll
  for (int i = 0; i < DHEAD; ++i) op[i] = av[i] * inv;
}

// ---------------------------------------------------------------------------
// out = LayerNorm(a + r) * g + b ; one 128-thread block per row, wave32 shfl.
// ---------------------------------------------------------------------------
__global__ void add_ln_kernel(const float* __restrict__ a,
                              const float* __restrict__ r,
                              const float* __restrict__ g,
                              const float* __restrict__ b,
                              float* __restrict__ out) {
  __shared__ float red[4];
  int row = blockIdx.x, tid = threadIdx.x;
  float v = a[(size_t)row * DMODEL + tid] + r[(size_t)row * DMODEL + tid];

  float s = v;
  for (int off = 16; off > 0; off >>= 1) s += __shfl_down(s, off);
  if ((tid & 31) == 0) red[tid >> 5] = s;
  __syncthreads();
  float mean = (red[0] + red[1] + red[2] + red[3]) * (1.0f / DMODEL);
  __syncthreads();

  float d = v - mean;
  float s2 = d * d;
  for (int off = 16; off > 0; off >>= 1) s2 += __shfl_down(s2, off);
  if ((tid & 31) == 0) red[tid >> 5] = s2;
  __syncthreads();
  float var = (red[0] + red[1] + red[2] + red[3]) * (1.0f / DMODEL);

  out[(size_t)row * DMODEL + tid] = d * rsqrtf(var + 1e-5f) * g[tid] + b[tid];
}

// ---------------------------------------------------------------------------
// Persistent pointer-network decoder: the entire 1023-step sequential loop
// in ONE workgroup (1024 threads = 32 wave32 on one WGP). State in LDS.
// All loop-invariant global operands are pre-transposed so every per-step
// global read is wave-coalesced:
//   wihT/whhT : [D, 4D]   (gate g at column g)
//   w1bT      : [D, D]    (output d at column d)
//   E1T       : [D, S]    (query s at column s)
// ---------------------------------------------------------------------------
__global__ void __launch_bounds__(1024, 1)
decoder_kernel(const float* __restrict__ enc,   // [S, D]
               const float* __restrict__ E1T,   // [D, S]  (no b1 inside)
               const float* __restrict__ wihT,  // [D, 4D]
               const float* __restrict__ whhT,  // [D, 4D]
               const float* __restrict__ bih,   // [4D]
               const float* __restrict__ bhh,   // [4D]
               const float* __restrict__ w1bT,  // [D, D]
               const float* __restrict__ pb1,   // [D]
               const float* __restrict__ pw2,   // [D]
               const float* __restrict__ pb2,   // [1]
               float* __restrict__ out) {       // [S-1, S]
  __shared__ float h[DMODEL], c[DMODEL], dec[DMODEL], hW[DMODEL];
  __shared__ float w2s[DMODEL];
  __shared__ float gates[4 * DMODEL];
  __shared__ unsigned maskw[S_LEN / 32];
  __shared__ float redv[32];
  __shared__ int   redi[32];
  __shared__ int   sidx;

  const int tid = threadIdx.x;
  if (tid < DMODEL) {
    h[tid] = 0.0f; c[tid] = 0.0f;
    dec[tid] = enc[tid];
    w2s[tid] = pw2[tid];
  }
  if (tid < S_LEN / 32) maskw[tid] = 0u;
  __syncthreads();

  const float b2 = pb2[0];

  for (int t = 0; t < S_LEN - 1; ++t) {
    // LSTMCell gates: g = dec@Wih^T + bih + h@Whh^T + bhh  (torch order i,f,g,o)
    if (tid < 4 * DMODEL) {
      float gacc = bih[tid] + bhh[tid];
      for (int j = 0; j < DMODEL; ++j)
        gacc += dec[j] * wihT[(size_t)j * (4 * DMODEL) + tid]
              + h[j]   * whhT[(size_t)j * (4 * DMODEL) + tid];
      gates[tid] = gacc;
    }
    __syncthreads();
    if (tid < DMODEL) {
      float ig = 1.0f / (1.0f + __expf(-gates[tid]));
      float fg = 1.0f / (1.0f + __expf(-gates[DMODEL + tid]));
      float gg = tanhf(gates[2 * DMODEL + tid]);
      float og = 1.0f / (1.0f + __expf(-gates[3 * DMODEL + tid]));
      float cn = fg * c[tid] + ig * gg;
      c[tid] = cn;
      h[tid] = og * tanhf(cn);
    }
    __syncthreads();
    // hW = b1 + h @ W1b^T   (coalesced via w1bT)
    if (tid < DMODEL) {
      float acc = pb1[tid];
      for (int j = 0; j < DMODEL; ++j) acc += h[j] * w1bT[(size_t)j * DMODEL + tid];
      hW[tid] = acc;
    }
    __syncthreads();

    // logit[s=tid] = b2 + relu(E1[s]+hW) . w2 ; mask -> -inf ; write + argmax
    {
      float lg = b2;
      for (int d = 0; d < DMODEL; ++d) {
        float v = E1T[(size_t)d * S_LEN + tid] + hW[d];   // coalesced
        lg += fmaxf(v, 0.0f) * w2s[d];
      }
      if ((maskw[tid >> 5] >> (tid & 31)) & 1u) lg = -__builtin_inff();
      out[(size_t)t * S_LEN + tid] = lg;

      float val = lg; int idx = tid;
      for (int off = 16; off > 0; off >>= 1) {
        float ov = __shfl_down(val, off);
        int   oi = __shfl_down(idx, off);
        if (ov > val || (ov == val && oi < idx)) { val = ov; idx = oi; }
      }
      if ((tid & 31) == 0) { redv[tid >> 5] = val; redi[tid >> 5] = idx; }
    }
    __syncthreads();
    if (tid == 0) {
      float bv = redv[0]; int bi = redi[0];
      for (int w = 1; w < 32; ++w)
        if (redv[w] > bv || (redv[w] == bv && redi[w] < bi)) { bv = redv[w]; bi = redi[w]; }
      maskw[bi >> 5] |= (1u << (bi & 31));
      sidx = bi;
    }
    __syncthreads();
    if (tid < DMODEL) dec[tid] = enc[(size_t)sidx * DMODEL + tid];
    __syncthreads();
  }
}

// ---------------------------------------------------------------------------
extern "C" void kernel_launch(void* const* d_in, const int* in_sizes, int n_in,
                              void* d_out, int out_size, void* d_ws, size_t ws_size,
                              hipStream_t stream) {
  (void)in_sizes; (void)n_in; (void)out_size; (void)ws_size;

  const float* x         = (const float*)d_in[0];
  const float* emb_w     = (const float*)d_in[1];
  const float* emb_b     = (const float*)d_in[2];
  const float* in_proj_w = (const float*)d_in[3];
  const float* in_proj_b = (const float*)d_in[4];
  const float* out_w     = (const float*)d_in[5];
  const float* out_b     = (const float*)d_in[6];
  const float* lin1_w    = (const float*)d_in[7];
  const float* lin1_b    = (const float*)d_in[8];
  const float* lin2_w    = (const float*)d_in[9];
  const float* lin2_b    = (const float*)d_in[10];
  const float* ln1_g     = (const float*)d_in[11];
  const float* ln1_b     = (const float*)d_in[12];
  const float* ln2_g     = (const float*)d_in[13];
  const float* ln2_b     = (const float*)d_in[14];
  const float* lstm_wih  = (const float*)d_in[15];
  const float* lstm_whh  = (const float*)d_in[16];
  const float* lstm_bih  = (const float*)d_in[17];
  const float* lstm_bhh  = (const float*)d_in[18];
  const float* ptr_w1    = (const float*)d_in[19];
  const float* ptr_b1    = (const float*)d_in[20];
  const float* ptr_w2    = (const float*)d_in[21];
  const float* ptr_b2    = (const float*)d_in[22];
  float* out = (float*)d_out;

  float* ws   = (float*)d_ws;
  float* xe   = ws;  ws += S_LEN * DMODEL;       // activations [S,D]
  float* qkv  = ws;  ws += S_LEN * 3 * DMODEL;   // [S,3D]
  float* ao   = ws;  ws += S_LEN * DMODEL;       // attention concat [S,D]
  float* prj  = ws;  ws += S_LEN * DMODEL;       // out-proj result
  float* x1   = ws;  ws += S_LEN * DMODEL;       // post-LN1
  float* ff1  = ws;  ws += S_LEN * DFF;          // [S,DFF]
  float* ff2  = ws;  ws += S_LEN * DMODEL;
  float* E1T  = ws;  ws += S_LEN * DMODEL;       // (enc @ W1a^T)^T  [D,S]
  float* wihT = ws;  ws += 4 * DMODEL * DMODEL;  // [D,4D]
  float* whhT = ws;  ws += 4 * DMODEL * DMODEL;  // [D,4D]
  float* w1bT = ws;  ws += DMODEL * DMODEL;      // [D,D]

  embed_kernel<<<S_LEN * DMODEL / 256, 256, 0, stream>>>(x, emb_w, emb_b, xe);

  // One-shot transposes of decoder loop invariants (coalesced per-step reads).
  transpose_kernel<<<(4 * DMODEL * DMODEL) / 256, 256, 0, stream>>>(
      lstm_wih, wihT, 4 * DMODEL, DMODEL, DMODEL, 0);
  transpose_kernel<<<(4 * DMODEL * DMODEL) / 256, 256, 0, stream>>>(
      lstm_whh, whhT, 4 * DMODEL, DMODEL, DMODEL, 0);
  transpose_kernel<<<(DMODEL * DMODEL) / 256, 256, 0, stream>>>(
      ptr_w1, w1bT, DMODEL, DMODEL, 2 * DMODEL, DMODEL);

  for (int l = 0; l < NLAYER; ++l) {
    // QKV projection [S,D] x [3D,D]^T
    gemm_wmma<<<dim3(3 * DMODEL / 16, S_LEN / 16), 32, 0, stream>>>(
        xe, DMODEL, in_proj_w + (size_t)l * 3 * DMODEL * DMODEL, DMODEL,
        in_proj_b + (size_t)l * 3 * DMODEL, qkv, 3 * DMODEL, DMODEL, 0, 0);
    // fused softmax attention
    attn_kernel<<<S_LEN * NHEAD / 256, 256, 0, stream>>>(qkv, ao);
    // output projection
    gemm_wmma<<<dim3(DMODEL / 16, S_LEN / 16), 32, 0, stream>>>(
        ao, DMODEL, out_w + (size_t)l * DMODEL * DMODEL, DMODEL,
        out_b + (size_t)l * DMODEL, prj, DMODEL, DMODEL, 0, 0);
    // x1 = LN(xe + proj)
    add_ln_kernel<<<S_LEN, DMODEL, 0, stream>>>(
        prj, xe, ln1_g + (size_t)l * DMODEL, ln1_b + (size_t)l * DMODEL, x1);
    // FFN up (relu fused)
    gemm_wmma<<<dim3(DFF / 16, S_LEN / 16), 32, 0, stream>>>(
        x1, DMODEL, lin1_w + (size_t)l * DFF * DMODEL, DMODEL,
        lin1_b + (size_t)l * DFF, ff1, DFF, DMODEL, 1, 0);
    // FFN down
    gemm_wmma<<<dim3(DMODEL / 16, S_LEN / 16), 32, 0, stream>>>(
        ff1, DFF, lin2_w + (size_t)l * DMODEL * DFF, DFF,
        lin2_b + (size_t)l * DMODEL, ff2, DMODEL, DFF, 0, 0);
    // xe = LN(x1 + ff2)
    add_ln_kernel<<<S_LEN, DMODEL, 0, stream>>>(
        ff2, x1, ln2_g + (size_t)l * DMODEL, ln2_b + (size_t)l * DMODEL, xe);
  }

  // E1T = (encoded @ W1a^T)^T stored [D,S] directly by the WMMA kernel
  // (W1a = first 128 cols of ptr_w1, row stride 2D; no bias — b1 lives in hW).
  gemm_wmma<<<dim3(DMODEL / 16, S_LEN / 16), 32, 0, stream>>>(
      xe, DMODEL, ptr_w1, 2 * DMODEL, nullptr, E1T, S_LEN, DMODEL, 0, 1);

  // entire sequential decode in one persistent workgroup
  decoder_kernel<<<1, 1024, 0, stream>>>(
      xe, E1T, wihT, whhT, lstm_bih, lstm_bhh,
      w1bT, ptr_b1, ptr_w2, ptr_b2, out);
}